// KernelEdges_32298154066197
// MI455X (gfx1250) — compile-verified
//
#include <hip/hip_runtime.h>
#include <hip/hip_bf16.h>

// MI455X (gfx1250) — store-bandwidth-bound fused Gram-kernel:
//   G = Xf·Xfᵀ  (Xf = X flattened [N, B*d]),  out[b,i,j] = exp((g_i+g_j-2G_ij)/σ²), diag=0
// Output = 134 MB stores (~5.8 µs @ 23.3 TB/s) dominates; GEMM is 4.3 GFLOP, so we
// use the exact-precision V_WMMA_F32_16X16X4_F32 path (no precision downgrade needed).

typedef __attribute__((ext_vector_type(2))) float v2f;
typedef __attribute__((ext_vector_type(8))) float v8f;

#define NN 2048
#define BATCH 8
#define DFEAT 64
#define KDIM (BATCH * DFEAT) // 512

// ---- Kernel 0: row squared-norms g[n] = sum_{b,f} X[b,n,f]^2  (8 KB into d_ws) ----
__global__ void ke_rownorm_kernel(const float* __restrict__ X, float* __restrict__ g) {
    int n = blockIdx.x * blockDim.x + threadIdx.x;
    if (n >= NN) return;
    float s = 0.0f;
#pragma unroll
    for (int b = 0; b < BATCH; ++b) {
        const float4* p = (const float4*)(X + (size_t)b * NN * DFEAT + (size_t)n * DFEAT);
#pragma unroll
        for (int f = 0; f < DFEAT / 4; ++f) {
            float4 v = p[f];
            s += v.x * v.x + v.y * v.y + v.z * v.z + v.w * v.w;
        }
    }
    g[n] = s;
}

// ---- Kernel 1: one wave32 per 64x64 tile of the symmetric 2048x2048 matrix.
// 4x4 grid of 16x16x4 F32 WMMA accumulators; fused exp-epilogue writes 8 batch copies.
__global__ __launch_bounds__(32) void ke_gram_kernel(const float* __restrict__ X,
                                                     const float* __restrict__ g,
                                                     const float* __restrict__ log_sigma,
                                                     float* __restrict__ out) {
    const int lane = threadIdx.x;      // 0..31 (wave32)
    const int row  = lane & 15;        // M (A-frag) / N (B-frag) within 16
    const int ps   = lane >> 4;        // K-pair select: lanes 16-31 hold K+2,K+3
    const int i0   = blockIdx.y * 64;
    const int j0   = blockIdx.x * 64;

    v8f acc[4][4];
#pragma unroll
    for (int mi = 0; mi < 4; ++mi)
#pragma unroll
        for (int ni = 0; ni < 4; ++ni)
            acc[mi][ni] = (v8f){0.f, 0.f, 0.f, 0.f, 0.f, 0.f, 0.f, 0.f};

    // Xf[n, k] with k = b*64 + f lives at X + b*NN*64 + n*64 + f.
    // K-step of 4: this lane loads (kb, kb+1) with kb = k0 + 2*ps — contiguous,
    // always inside one 64-float feature chunk since kb is even -> one b64 load.
    for (int k0 = 0; k0 < KDIM; k0 += 4) {
        const int kb = k0 + 2 * ps;
        const float* base = X + (size_t)(kb >> 6) * (NN * DFEAT) + (kb & 63);

        v2f afr[4], bfr[4];
#pragma unroll
        for (int mi = 0; mi < 4; ++mi)
            afr[mi] = *(const v2f*)(base + (size_t)(i0 + 16 * mi + row) * DFEAT);
#pragma unroll
        for (int ni = 0; ni < 4; ++ni)
            bfr[ni] = *(const v2f*)(base + (size_t)(j0 + 16 * ni + row) * DFEAT);

#pragma unroll
        for (int mi = 0; mi < 4; ++mi)
#pragma unroll
            for (int ni = 0; ni < 4; ++ni)
                acc[mi][ni] = __builtin_amdgcn_wmma_f32_16x16x4_f32(
                    /*neg_a=*/false, afr[mi], /*neg_b=*/false, bfr[ni],
                    /*c_mod=*/(short)0, acc[mi][ni],
                    /*reuse_a=*/false, /*reuse_b=*/false);
    }

    // Epilogue: sq = max(g_i + g_j - 2G, 0);  A = exp(sq * e^{-2 log_sigma}); diag -> 0.
    const float inv_s2 = __expf(-2.0f * log_sigma[0]);

    float gcol[4];
#pragma unroll
    for (int ni = 0; ni < 4; ++ni)
        gcol[ni] = g[j0 + 16 * ni + row];

    float grow[4][8];
#pragma unroll
    for (int mi = 0; mi < 4; ++mi)
#pragma unroll
        for (int r = 0; r < 8; ++r)
            grow[mi][r] = g[i0 + 16 * mi + 8 * ps + r];

    // C/D layout: VGPR r -> rows r (lanes 0-15) and r+8 (lanes 16-31); col = lane&15.
#pragma unroll
    for (int mi = 0; mi < 4; ++mi) {
#pragma unroll
        for (int r = 0; r < 8; ++r) {
            const int gr = i0 + 16 * mi + 8 * ps + r;
            const size_t rowbase = (size_t)gr * NN;
#pragma unroll
            for (int ni = 0; ni < 4; ++ni) {
                const int gc = j0 + 16 * ni + row;
                float G2 = acc[mi][ni][r];
                float sq = fmaxf(grow[mi][r] + gcol[ni] - 2.0f * G2, 0.0f);
                float v  = __expf(sq * inv_s2);
                if (gr == gc) v = 0.0f;
                const size_t off = rowbase + (size_t)gc;
#pragma unroll
                for (int b = 0; b < BATCH; ++b)
                    out[(size_t)b * (size_t)NN * NN + off] = v;
            }
        }
    }
}

extern "C" void kernel_launch(void* const* d_in, const int* in_sizes, int n_in,
                              void* d_out, int out_size, void* d_ws, size_t ws_size,
                              hipStream_t stream) {
    const float* X         = (const float*)d_in[0];
    const float* log_sigma = (const float*)d_in[1];
    float* out             = (float*)d_out;
    float* g               = (float*)d_ws; // 2048 floats = 8 KB

    ke_rownorm_kernel<<<NN / 256, 256, 0, stream>>>(X, g);

    dim3 grid(NN / 64, NN / 64); // 32 x 32 tiles, one wave32 each
    ke_gram_kernel<<<grid, 32, 0, stream>>>(X, g, log_sigma, out);
}